// SubwordTokenizerLayer_77541339562498
// MI455X (gfx1250) — compile-verified
//
#include <hip/hip_runtime.h>
#include <stdint.h>

// ---------------------------------------------------------------------------
// FNV-1a subword hashing (fastText GetSubwordIndex), MI455X / gfx1250.
// Pure HBM-streaming problem (~185 MB total => ~8us at 23.3 TB/s).
// CDNA5 paths used: async global->LDS (ASYNCcnt) double-buffered pipeline,
// s_wait_asynccnt, split barriers, NT temporal hints, global_prefetch_b8.
// ---------------------------------------------------------------------------

typedef __attribute__((ext_vector_type(4))) unsigned int uv4;
typedef __attribute__((ext_vector_type(4))) int          iv4;
// Builtin signature (from clang diagnostic): arg0 is int4* in AS(1) (__device__),
// arg1 is the LDS-side pointer; model both with explicit address spaces.
typedef __attribute__((address_space(1))) iv4 as1_iv4;
typedef __attribute__((address_space(3))) iv4 as3_iv4;

static constexpr unsigned kFnvPrime = 16777619u;
static constexpr unsigned kFnvInit  = 2166136261u;
static constexpr unsigned kBucket   = 2000000u;
static constexpr unsigned kVocab    = 2000000u;

#define BLOCK       256
#define TILE        256                 // subwords per tile
#define TILE_BYTES  (TILE * 32)         // 8 KB of byte data per tile

#if defined(__gfx1250__) && \
    __has_builtin(__builtin_amdgcn_global_load_async_to_lds_b128) && \
    __has_builtin(__builtin_amdgcn_s_wait_asynccnt)
#define USE_ASYNC_LDS 1
#else
#define USE_ASYNC_LDS 0
#warning "gfx1250 async-to-LDS builtins unavailable on this toolchain; falling back to direct NT b128 loads"
#endif

__device__ __forceinline__ unsigned fnv8(uv4 b0, uv4 b1, unsigned len) {
  // Matches reference: h = where(j < len, (h ^ byte) * prime, h), j = 0..7
  unsigned h = kFnvInit;
  h = (0u < len) ? (h ^ b0.x) * kFnvPrime : h;
  h = (1u < len) ? (h ^ b0.y) * kFnvPrime : h;
  h = (2u < len) ? (h ^ b0.z) * kFnvPrime : h;
  h = (3u < len) ? (h ^ b0.w) * kFnvPrime : h;
  h = (4u < len) ? (h ^ b1.x) * kFnvPrime : h;
  h = (5u < len) ? (h ^ b1.y) * kFnvPrime : h;
  h = (6u < len) ? (h ^ b1.z) * kFnvPrime : h;
  h = (7u < len) ? (h ^ b1.w) * kFnvPrime : h;
  return h;
}

__global__ __launch_bounds__(BLOCK) void fnv_tiles_kernel(
    const int* __restrict__ bytes,   // [N, 8] int32, each holds one UTF-8 byte
    const int* __restrict__ lens,    // [N]
    long long* __restrict__ out,     // [N] int64
    int numTiles)
{
  const int tid = (int)threadIdx.x;

#if USE_ASYNC_LDS
  __shared__ uv4 lds[2][TILE * 2];   // double buffer, 8 KB each

  const int stride = (int)gridDim.x;

  auto stage = [&](int t, int buf) {
    // Each lane DMAs 2 x 16B; whole block stages the 8 KB tile linearly.
    const char* g = (const char*)bytes + (size_t)t * TILE_BYTES + (size_t)tid * 16;
    unsigned l0 = (unsigned)(uintptr_t)&lds[buf][tid];
    unsigned l1 = (unsigned)(uintptr_t)&lds[buf][TILE + tid];
    __builtin_amdgcn_global_load_async_to_lds_b128(
        (as1_iv4*)(uintptr_t)g, (as3_iv4*)l0, 0, 0);
    __builtin_amdgcn_global_load_async_to_lds_b128(
        (as1_iv4*)(uintptr_t)(g + TILE_BYTES / 2), (as3_iv4*)l1, 0, 0);
  };

  int buf = 0;
  const int t0 = (int)blockIdx.x;
  if (t0 < numTiles) stage(t0, 0);

  for (int t = t0; t < numTiles; t += stride, buf ^= 1) {
    const int tn = t + stride;
    if (tn < numTiles) {
      stage(tn, buf ^ 1);                           // prefetch next tile -> other buffer
      __builtin_prefetch(lens + (size_t)tn * TILE + tid, 0, 0);
      __builtin_amdgcn_s_wait_asynccnt(2);          // tile t's 2 transfers done (per wave)
    } else {
      __builtin_amdgcn_s_wait_asynccnt(0);
    }
    __syncthreads();                                // all waves' portions of tile t in LDS

    const int s = t * TILE + tid;
    uv4 b0 = lds[buf][2 * tid + 0];
    uv4 b1 = lds[buf][2 * tid + 1];
    unsigned len = (unsigned)__builtin_nontemporal_load(lens + s);
    unsigned idx = fnv8(b0, b1, len) % kBucket + kVocab;
    __builtin_nontemporal_store((long long)(unsigned long long)idx, out + s);

    __syncthreads();                                // buffer reusable for tile t+2's stage
  }
#else
  // Direct-path fallback: two NT b128 loads + NT b32 length load per subword.
  const uv4* __restrict__ bv = (const uv4*)bytes;
  const int n = numTiles * TILE;
  for (int s = (int)(blockIdx.x * BLOCK) + tid; s < n;
       s += (int)(gridDim.x * BLOCK)) {
    uv4 b0 = __builtin_nontemporal_load(bv + 2 * (size_t)s + 0);
    uv4 b1 = __builtin_nontemporal_load(bv + 2 * (size_t)s + 1);
    unsigned len = (unsigned)__builtin_nontemporal_load(lens + s);
    unsigned idx = fnv8(b0, b1, len) % kBucket + kVocab;
    __builtin_nontemporal_store((long long)(unsigned long long)idx, out + s);
  }
#endif
}

// Scalar tail for N not a multiple of TILE (N = 2^22 in practice, so unused,
// but keeps the launch correct for any N).
__global__ void fnv_tail_kernel(const int* __restrict__ bytes,
                                const int* __restrict__ lens,
                                long long* __restrict__ out,
                                int start, int n)
{
  int s = start + (int)(blockIdx.x * blockDim.x + threadIdx.x);
  if (s < n) {
    const uv4* bv = (const uv4*)bytes;
    uv4 b0 = bv[2 * (size_t)s + 0];
    uv4 b1 = bv[2 * (size_t)s + 1];
    unsigned idx = fnv8(b0, b1, (unsigned)lens[s]) % kBucket + kVocab;
    out[s] = (long long)(unsigned long long)idx;
  }
}

extern "C" void kernel_launch(void* const* d_in, const int* in_sizes, int n_in,
                              void* d_out, int out_size, void* d_ws, size_t ws_size,
                              hipStream_t stream) {
  (void)d_ws; (void)ws_size; (void)out_size;
  const int* bytes = (const int*)d_in[0];                 // [N, 8] int32
  const int* lens  = (const int*)d_in[1];                 // [N]    int32
  long long* out   = (long long*)d_out;                   // [N]    int64

  const int N = (n_in > 1) ? in_sizes[1] : (in_sizes[0] / 8);
  const int numTiles = N / TILE;
  const int rem = N - numTiles * TILE;

  if (numTiles > 0) {
    // 16384 tiles for N=4M; 4096 blocks => 4 tiles per block, double-buffered.
    int grid = numTiles < 4096 ? numTiles : 4096;
    fnv_tiles_kernel<<<grid, BLOCK, 0, stream>>>(bytes, lens, out, numTiles);
  }
  if (rem > 0) {
    fnv_tail_kernel<<<(rem + BLOCK - 1) / BLOCK, BLOCK, 0, stream>>>(
        bytes, lens, out, numTiles * TILE, N);
  }
}